// AnomalyAttention_41712722378782
// MI455X (gfx1250) — compile-verified
//
#include <hip/hip_runtime.h>
#include <math.h>

typedef __attribute__((ext_vector_type(2))) float v2f;
typedef __attribute__((ext_vector_type(8))) float v8f;

constexpr int Bc = 4, Lc = 2048, Hc = 8, Ec = 64;
constexpr int BH   = Bc * Hc;               // 32 batch*head slices
constexpr int RT   = Lc / 16;               // 128 row tiles
constexpr int NTRI = RT * (RT + 1) / 2;     // 8256 lower-triangular tiles

__device__ __forceinline__ v8f wmma_f32_k4(v2f a, v2f b, v8f c) {
  // D = A(16x4 f32) * B(4x16 f32) + C(16x16 f32)
  return __builtin_amdgcn_wmma_f32_16x16x4_f32(false, a, false, b, (short)0, c,
                                               false, false);
}

// ---------------------------------------------------------------------------
// Kernel 1: scores = scale * Q K^T, lower-triangular 16x16 tiles only.
// One wave per tile; 16 WMMA (K=4) per tile for E=64.
// ---------------------------------------------------------------------------
__global__ void __launch_bounds__(128)
scores_qkt(const float* __restrict__ Q, const float* __restrict__ K,
           float* __restrict__ series) {
  const int lane = threadIdx.x & 31;
  const int t    = blockIdx.x * 4 + (threadIdx.x >> 5);  // triangular tile id
  const int bh   = blockIdx.y;
  const int b    = bh >> 3, h = bh & 7;

  // decode triangular index t -> (row tile r, col tile ct), ct <= r
  int r = (int)((sqrtf(8.0f * (float)t + 1.0f) - 1.0f) * 0.5f);
  while ((r + 1) * (r + 2) / 2 <= t) ++r;
  while (r * (r + 1) / 2 > t) --r;
  const int ct = t - r * (r + 1) / 2;

  const int l0 = r << 4, s0 = ct << 4;
  const int m  = lane & 15;           // A row (also B column index)
  const int kb = (lane >> 4) << 1;    // K sub-offset {0,2} per half-wave

  const float* qrow = Q + (((size_t)b * Lc + (l0 + m)) * Hc + h) * Ec;
  const float* krow = K + (((size_t)b * Lc + (s0 + m)) * Hc + h) * Ec;

  v8f acc = {0.f, 0.f, 0.f, 0.f, 0.f, 0.f, 0.f, 0.f};
#pragma unroll
  for (int kc = 0; kc < Ec; kc += 4) {
    const v2f a  = *(const v2f*)(qrow + kc + kb);   // Q[l0+m, kc+kb .. +1]
    const v2f bb = *(const v2f*)(krow + kc + kb);   // K[s0+m, kc+kb .. +1]
    acc = wmma_f32_k4(a, bb, acc);
  }

  // D layout: VGPR rr -> M = rr + 8*(lane>=16), N = lane&15
  const float scale = 0.125f;  // 1/sqrt(64)
  float* orow = series +
      ((size_t)bh * Lc + (l0 + ((lane >> 4) << 3))) * Lc + s0 + m;
#pragma unroll
  for (int rr = 0; rr < 8; ++rr)
    orow[(size_t)rr * Lc] = acc[rr] * scale;
}

// ---------------------------------------------------------------------------
// Kernel 2: causal row softmax in place + prior/sig rows.
// One 256-thread block per (bh, l) row.
// ---------------------------------------------------------------------------
__global__ void __launch_bounds__(256)
softmax_prior(const float* __restrict__ sigma,
              float* __restrict__ series,
              float* __restrict__ prior,
              float* __restrict__ sigb) {
  __shared__ float red[256];
  const int row = blockIdx.x;          // bh*L + l
  const int bh  = row >> 11;           // L = 2048
  const int l   = row & (Lc - 1);
  const int b   = bh >> 3, h = bh & 7;
  const int tid = threadIdx.x;

  float* srow = series + (size_t)row * Lc;
  const int nv = l + 1;                // valid causal length

  // row max
  float mx = -3.0e38f;
  for (int s = tid; s < nv; s += 256) mx = fmaxf(mx, srow[s]);
  red[tid] = mx;
  __syncthreads();
  for (int o = 128; o > 0; o >>= 1) {
    if (tid < o) red[tid] = fmaxf(red[tid], red[tid + o]);
    __syncthreads();
  }
  mx = red[0];
  __syncthreads();

  // exp + sum
  float sum = 0.f;
  for (int s = tid; s < nv; s += 256) {
    const float e = __expf(srow[s] - mx);
    srow[s] = e;
    sum += e;
  }
  red[tid] = sum;
  __syncthreads();
  for (int o = 128; o > 0; o >>= 1) {
    if (tid < o) red[tid] += red[tid + o];
    __syncthreads();
  }
  const float inv = 1.0f / red[0];
  for (int s = tid; s < nv; s += 256) srow[s] *= inv;
  for (int s = nv + tid; s < Lc; s += 256) srow[s] = 0.f;  // masked -> 0

  // sig = 3^(sigmoid(5x) + 1e-5) - 1 ; prior = gaussian(|l-s|, sig)
  const float x  = sigma[((size_t)b * Lc + l) * Hc + h];
  float sg = 1.0f / (1.0f + __expf(-5.0f * x)) + 1e-5f;
  sg = __expf(1.0986122886681098f * sg) - 1.0f;   // 3^sg - 1
  const float coef = 0.3989422804014327f / sg;     // 1/(sqrt(2pi)*sig)
  const float qexp = -0.5f / (sg * sg);
  float* prow = prior + (size_t)row * Lc;
  float* grow = sigb + (size_t)row * Lc;
  for (int s = tid; s < Lc; s += 256) {
    const float d = (float)(l - s);
    prow[s] = coef * __expf(d * d * qexp);
    grow[s] = sg;
  }
}

// ---------------------------------------------------------------------------
// Kernel 3: V = series @ values. One wave per 16-row tile; 4 accumulators
// cover the full E=64, sharing one A-frag per K-step. Causal K-depth only.
// ---------------------------------------------------------------------------
__global__ void __launch_bounds__(32)
attn_v(const float* __restrict__ series, const float* __restrict__ Vin,
       float* __restrict__ Vout) {
  const int lane = threadIdx.x & 31;
  const int rt   = blockIdx.x;       // row tile
  const int bh   = blockIdx.y;
  const int b    = bh >> 3, h = bh & 7;
  const int l0   = rt << 4;
  const int m    = lane & 15;
  const int kb   = (lane >> 4) << 1;

  const float* arow = series + ((size_t)bh * Lc + (l0 + m)) * Lc;

  v8f acc0 = {0.f, 0.f, 0.f, 0.f, 0.f, 0.f, 0.f, 0.f};
  v8f acc1 = acc0, acc2 = acc0, acc3 = acc0;

  const int ksteps = (rt + 1) << 2;  // K=4 steps up to causal boundary
  for (int ks = 0; ks < ksteps; ++ks) {
    const int s = (ks << 2) + kb;
    __builtin_prefetch(arow + s + 128, 0, 0);      // stream series ahead
    const v2f a = *(const v2f*)(arow + s);         // series[l0+m, s..s+1]
    const float* v0 = Vin + (((size_t)b * Lc + s) * Hc + h) * Ec + m;
    const float* v1 = v0 + Hc * Ec;                // next s
    v2f b0, b1, b2, b3;
    b0[0] = v0[0];  b0[1] = v1[0];
    b1[0] = v0[16]; b1[1] = v1[16];
    b2[0] = v0[32]; b2[1] = v1[32];
    b3[0] = v0[48]; b3[1] = v1[48];
    acc0 = wmma_f32_k4(a, b0, acc0);
    acc1 = wmma_f32_k4(a, b1, acc1);
    acc2 = wmma_f32_k4(a, b2, acc2);
    acc3 = wmma_f32_k4(a, b3, acc3);
  }

  const int mh = (lane >> 4) << 3;
#pragma unroll
  for (int rr = 0; rr < 8; ++rr) {
    float* o = Vout + (((size_t)b * Lc + (l0 + mh + rr)) * Hc + h) * Ec + m;
    o[0]  = acc0[rr];
    o[16] = acc1[rr];
    o[32] = acc2[rr];
    o[48] = acc3[rr];
  }
}

// ---------------------------------------------------------------------------
extern "C" void kernel_launch(void* const* d_in, const int* in_sizes, int n_in,
                              void* d_out, int out_size, void* d_ws,
                              size_t ws_size, hipStream_t stream) {
  const float* Q     = (const float*)d_in[0];
  const float* K     = (const float*)d_in[1];
  const float* V     = (const float*)d_in[2];
  const float* sigma = (const float*)d_in[3];

  float* out    = (float*)d_out;
  float* Vout   = out;                                      // [B,L,H,E]
  float* series = out + (size_t)Bc * Lc * Hc * Ec;          // [B,H,L,L]
  float* prior  = series + (size_t)BH * Lc * Lc;            // [B,H,L,L]
  float* sigb   = prior + (size_t)BH * Lc * Lc;             // [B,H,L,L]

  // 1) lower-triangular scaled scores via WMMA (4 waves / block)
  scores_qkt<<<dim3(NTRI / 4, BH), 128, 0, stream>>>(Q, K, series);
  // 2) causal softmax in place + prior + sig rows
  softmax_prior<<<dim3(BH * Lc), 256, 0, stream>>>(sigma, series, prior, sigb);
  // 3) V = series @ values via WMMA
  attn_v<<<dim3(RT, BH), 32, 0, stream>>>(series, V, Vout);
}